// similarLoss_76854144794999
// MI455X (gfx1250) — compile-verified
//
#include <hip/hip_runtime.h>

typedef __attribute__((ext_vector_type(2))) float v2f;
typedef __attribute__((ext_vector_type(8))) float v8f;

// Exact full-wave32 sum using V_WMMA_F32_16X16X4_F32.
// A (16x4 f32): VGPR0 = lane acc (lanes 0-15 -> K=0, lanes 16-31 -> K=2),
//               VGPR1 = 0 (K=1,3). B = ones (4x16).
// D[m][n] = acc[m] + acc[m+16]; each lane holds 8 rows of one column, so
// sum(c[0..7]) is one half-wave's total; shfl_xor(16) adds the other half.
static __device__ __forceinline__ float wave_sum_wmma(float acc) {
    v2f a; a[0] = acc;  a[1] = 0.0f;
    v2f b; b[0] = 1.0f; b[1] = 1.0f;
    v8f c = {};
    c = __builtin_amdgcn_wmma_f32_16x16x4_f32(
        /*neg_a=*/false, a, /*neg_b=*/false, b,
        /*c_mod=*/(short)0, c, /*reuse_a=*/false, /*reuse_b=*/false);
    float s = ((c[0] + c[1]) + (c[2] + c[3])) + ((c[4] + c[5]) + (c[6] + c[7]));
    s += __shfl_xor(s, 16, 32);
    return s;
}

// Border-coverage weight for a 9-wide zero-padded box window, N=256:
// c(x) = 9 - max(0,4-x) - max(0, x-251)
static __device__ __forceinline__ float cov256(int x) {
    return 9.0f - (float)max(0, 4 - x) - (float)max(0, x - 251);
}

__global__ void wssd_partial_kernel(const float4* __restrict__ real4,
                                    const float4* __restrict__ fake4,
                                    float* __restrict__ partial,
                                    int n4) {
    constexpr int W4 = 64;   // W(=256) / 4
    constexpr int H  = 256;
    const float inv81 = 1.0f / 81.0f;

    float acc = 0.0f;
    const int stride = gridDim.x * blockDim.x;
    for (int i = blockIdx.x * blockDim.x + threadIdx.x; i < n4; i += stride) {
        float4 r = real4[i];
        float4 f = fake4[i];
        int w0 = (i % W4) * 4;        // four consecutive w positions, same h
        int h  = (i / W4) % H;

        float ch  = cov256(h);
        float cw0 = cov256(w0);
        float cw1 = cov256(w0 + 1);
        float cw2 = cov256(w0 + 2);
        float cw3 = cov256(w0 + 3);

        float d0 = r.x - f.x;
        float d1 = r.y - f.y;
        float d2 = r.z - f.z;
        float d3 = r.w - f.w;

        float s = d0 * d0 * cw0 + d1 * d1 * cw1 + d2 * d2 * cw2 + d3 * d3 * cw3;
        acc = fmaf(s, ch * inv81, acc);
    }

    // All threads converged here with equal trip counts -> EXEC all 1s.
    float ws = wave_sum_wmma(acc);

    __shared__ float lds[8];
    const int lane = threadIdx.x & 31;
    const int wave = threadIdx.x >> 5;
    if (lane == 0) lds[wave] = ws;
    __syncthreads();
    if (threadIdx.x == 0) {
        float t = 0.0f;
        const int nw = blockDim.x >> 5;
        for (int k = 0; k < nw; ++k) t += lds[k];
        partial[blockIdx.x] = t;
    }
}

__global__ void wssd_final_kernel(const float* __restrict__ partial,
                                  int n_partial,
                                  float* __restrict__ out,
                                  float inv_total) {
    float acc = 0.0f;
    // n_partial (2048) is a multiple of blockDim.x (256): uniform trip count.
    for (int i = threadIdx.x; i < n_partial; i += blockDim.x) acc += partial[i];

    float ws = wave_sum_wmma(acc);

    __shared__ float lds[8];
    const int lane = threadIdx.x & 31;
    const int wave = threadIdx.x >> 5;
    if (lane == 0) lds[wave] = ws;
    __syncthreads();
    if (threadIdx.x == 0) {
        float t = 0.0f;
        const int nw = blockDim.x >> 5;
        for (int k = 0; k < nw; ++k) t += lds[k];
        out[0] = t * inv_total;
    }
}

extern "C" void kernel_launch(void* const* d_in, const int* in_sizes, int n_in,
                              void* d_out, int out_size, void* d_ws, size_t ws_size,
                              hipStream_t stream) {
    const float* real = (const float*)d_in[0];
    const float* fake = (const float*)d_in[1];
    float* out = (float*)d_out;
    float* partial = (float*)d_ws;

    const int total = in_sizes[0];          // 2*1*64*256*256 = 8,388,608
    const int n4 = total / 4;               // float4 elements
    const int THREADS = 256;
    const int BLOCKS = 2048;                // 8 KB of partials in d_ws

    wssd_partial_kernel<<<BLOCKS, THREADS, 0, stream>>>(
        (const float4*)real, (const float4*)fake, partial, n4);
    wssd_final_kernel<<<1, THREADS, 0, stream>>>(
        partial, BLOCKS, out, 1.0f / (float)total);
}